// Phi_r_32384053412565
// MI455X (gfx1250) — compile-verified
//
#include <hip/hip_runtime.h>
#include <math.h>

typedef float v2f __attribute__((ext_vector_type(2)));
typedef float v8f __attribute__((ext_vector_type(8)));

#define N_T   5
#define NN    1024              // N = 32*32
#define NB    2
#define NMAT  8                 // NB*(N_T-1)
#define MATS  (NN*NN)           // 1M elems per matrix
#define QDIM  (N_T*NN)          // 5120

// workspace layout in floats
static const size_t WS_COEF = 0;                       // 8*1024*9 = 73728
static const size_t WS_A    = (size_t)1 << 17;         // 131072, 8 matrices (later overwritten with E blocks)
static const size_t WS_M2   = WS_A  + (size_t)NMAT * MATS;
static const size_t WS_G    = WS_M2 + (size_t)NMAT * MATS;

__device__ __forceinline__ float softplus10(float x) {
    float z = 10.f * x;
    return (z > 20.f) ? x : log1pf(expf(z)) * 0.1f;
}

// ---------------------------------------------------------------------------
// K0: decode params -> 9 stencil coefficients per (bt, row).
// Faithful to decode_param_CNN's raw reshapes:
//   kappa/m use flat index j = n*5+t over the (5,32,32) channel block,
//   gamma/vx/vy use channel (base+t) at spatial n.
// ---------------------------------------------------------------------------
__global__ void build_coef_kernel(const float* __restrict__ p, float* __restrict__ coef) {
    int gid = blockIdx.x * blockDim.x + threadIdx.x;   // NMAT*NN = 8192
    if (gid >= NMAT * NN) return;
    int bt = gid / NN;
    int n  = gid % NN;
    int b  = bt / (N_T - 1);
    int t  = bt % (N_T - 1) + 1;                       // 1..4
    const float* pb = p + (size_t)b * 7 * N_T * NN;

    int j   = n * N_T + t;
    int ch  = j >> 10;
    int rem = j & 1023;
    float kap = softplus10(pb[(0 * N_T + ch) * NN + rem]);
    float m1  =            pb[(1 * N_T + ch) * NN + rem];
    float m2  =            pb[(2 * N_T + ch) * NN + rem];
    float gam = softplus10(pb[(3 * N_T + t) * NN + n]);
    float vx  =            pb[(4 * N_T + t) * NN + n];
    float vy  =            pb[(5 * N_T + t) * NN + n];

    float H11 = gam + vx * vx;
    float H22 = gam + vy * vy;
    float H12 = vx * vy;                               // 0.5*(H01+H10)

    float* c = coef + (size_t)gid * 9;                 // index k = (dy+1)*3 + (dx+1)
    c[0] = -0.5f * H12;                                // (-1,-1)
    c[1] = -H22 - 0.5f * m2;                           // ( 0,-1)
    c[2] =  0.5f * H12;                                // (+1,-1)
    c[3] = -H11 - 0.5f * m1;                           // (-1, 0)
    c[4] = kap * kap + 2.f * H11 + 2.f * H22;          // center
    c[5] = -H11 + 0.5f * m1;                           // (+1, 0)
    c[6] =  0.5f * H12;                                // (-1,+1)
    c[7] = -H22 + 0.5f * m2;                           // ( 0,+1)
    c[8] = -0.5f * H12;                                // (+1,+1)
}

// ---------------------------------------------------------------------------
// K1: materialize dense A (coefficients depend only on the row).
// ---------------------------------------------------------------------------
__global__ void build_A_kernel(const float* __restrict__ coef, float* __restrict__ Aout) {
    unsigned gid = blockIdx.x * blockDim.x + threadIdx.x;      // 8M
    if (gid >= (unsigned)NMAT * MATS) return;
    int mat = gid >> 20;
    int e   = gid & (MATS - 1);
    int r = e >> 10, c = e & 1023;
    int xr = r & 31, yr = r >> 5;
    int xc = c & 31, yc = c >> 5;
    int dx = xc - xr, dy = yc - yr;
    float v = 0.f;
    if (dx >= -1 && dx <= 1 && dy >= -1 && dy <= 1)
        v = coef[((size_t)mat * NN + r) * 9 + (dy + 1) * 3 + (dx + 1)];
    Aout[gid] = v;
}

// ---------------------------------------------------------------------------
// K2: fp32 WMMA GEMM.  z < 8 : M2[z] = (A[z]+I)(A[z]+I)
//                      z >= 8: G[z-8] = B0^T B0   (B0 = A[(z-8)*4], no +I)
// block = 256 thr = 8 waves, tile 128x128, K-step 32, wave patch 32x64.
// Both LDS tiles are stored row-per-matrix-row/col with K contiguous, so every
// WMMA operand fragment is one aligned 64-bit DS load (no register repacking).
// ---------------------------------------------------------------------------
#define BM 128
#define BN 128
#define BK 32
#define LDK (BK + 4)

__global__ __launch_bounds__(256) void gemm_wmma_kernel(float* __restrict__ ws) {
    const int mat = blockIdx.z;
    const float* L;
    const float* R;
    float* D;
    bool addI, transL;
    if (mat < NMAT) {
        L = ws + WS_A + (size_t)mat * MATS; R = L;
        D = ws + WS_M2 + (size_t)mat * MATS;
        addI = true;  transL = false;
    } else {
        int b = mat - NMAT;
        L = ws + WS_A + (size_t)(b * 4) * MATS; R = L;
        D = ws + WS_G + (size_t)b * MATS;
        addI = false; transL = true;
    }

    __shared__ float As[BM][LDK];     // As[m][k]
    __shared__ float Bs[BN][LDK];     // Bs[n][k]  (K contiguous for b64 frag loads)

    const int tid  = threadIdx.x;
    const int wave = tid >> 5;
    const int lane = tid & 31;
    const int half = lane >> 4;
    const int lc   = lane & 15;
    const int wm   = (wave >> 1) * 32;     // 0,32,64,96
    const int wn   = (wave & 1) * 64;      // 0,64
    const int rowBase = blockIdx.x * BM;
    const int colBase = blockIdx.y * BN;

    const v8f vzero = {0.f,0.f,0.f,0.f,0.f,0.f,0.f,0.f};
    v8f acc[2][4];
#pragma unroll
    for (int im = 0; im < 2; ++im)
#pragma unroll
        for (int in = 0; in < 4; ++in) acc[im][in] = vzero;

    for (int k0 = 0; k0 < NN; k0 += BK) {
        __syncthreads();
        if (!transL) {
            int m  = tid >> 1;
            int ks = (tid & 1) * 16;
            int gm = rowBase + m;
#pragma unroll
            for (int i = 0; i < 16; ++i) {
                int gk = k0 + ks + i;
                float v = L[(size_t)gm * NN + gk];
                if (addI && gm == gk) v += 1.f;
                As[m][ks + i] = v;
            }
        } else {                             // As[m][k] = L[k][m], coalesced over m
            int k  = tid >> 3;
            int ms = (tid & 7) * 16;
            int gk = k0 + k;
#pragma unroll
            for (int i = 0; i < 16; ++i)
                As[ms + i][k] = L[(size_t)gk * NN + rowBase + ms + i];
        }
        {
            int kr = tid >> 3;
            int ns = (tid & 7) * 16;
            int gk = k0 + kr;
#pragma unroll
            for (int i = 0; i < 16; ++i) {   // coalesced global read, transposed LDS write
                int gn = colBase + ns + i;
                float v = R[(size_t)gk * NN + gn];
                if (addI && gk == gn) v += 1.f;
                Bs[ns + i][kr] = v;
            }
        }
        __syncthreads();

#pragma unroll
        for (int kk = 0; kk < BK; kk += 4) {
            // A 16x4 f32: VGPR0 = K(2*half), VGPR1 = K(2*half+1)  -> contiguous pair
            // B 4x16 f32: lanes 0-15 K=0..1, lanes 16-31 K=2..3   -> contiguous pair
            v2f afrag[2], bfrag[4];
#pragma unroll
            for (int im = 0; im < 2; ++im) {
                int m = wm + im * 16 + lc;
                afrag[im] = *(const v2f*)&As[m][kk + 2 * half];
            }
#pragma unroll
            for (int in = 0; in < 4; ++in) {
                int n = wn + in * 16 + lc;
                bfrag[in] = *(const v2f*)&Bs[n][kk + 2 * half];
            }
#pragma unroll
            for (int im = 0; im < 2; ++im)
#pragma unroll
                for (int in = 0; in < 4; ++in)
                    acc[im][in] = __builtin_amdgcn_wmma_f32_16x16x4_f32(
                        false, afrag[im], false, bfrag[in],
                        (short)0, acc[im][in], false, false);
        }
    }

#pragma unroll
    for (int im = 0; im < 2; ++im)
#pragma unroll
        for (int in = 0; in < 4; ++in) {
            int col = colBase + wn + in * 16 + lc;
#pragma unroll
            for (int v = 0; v < 8; ++v) {
                int row = rowBase + wm + im * 16 + v + half * 8;
                D[(size_t)row * NN + col] = acc[im][in][v];
            }
        }
}

// ---------------------------------------------------------------------------
// K3: in-place tile-pair transform  M <- alpha*(M + M^T) + beta*I
//     (alpha=0.5,beta=0  : symmetrize M2;  alpha=-0.5,beta=-1 : A -> E blocks)
// ---------------------------------------------------------------------------
__global__ __launch_bounds__(256) void sym_kernel(float* __restrict__ Mbase, float alpha, float beta) {
    int p = blockIdx.x;                    // 0..527 (32*33/2 tile pairs of 32x32)
    int ti = 0;
    while (p >= 32 - ti) { p -= 32 - ti; ++ti; }
    int tj = ti + p;
    float* M = Mbase + (size_t)blockIdx.y * MATS;

    __shared__ float T1[32][33];
    __shared__ float T2[32][33];
    int r  = threadIdx.x >> 3;
    int c4 = (threadIdx.x & 7) * 4;
#pragma unroll
    for (int i = 0; i < 4; ++i) {
        T1[r][c4 + i] = M[(size_t)(ti * 32 + r) * NN + tj * 32 + c4 + i];
        T2[r][c4 + i] = M[(size_t)(tj * 32 + r) * NN + ti * 32 + c4 + i];
    }
    __syncthreads();
#pragma unroll
    for (int i = 0; i < 4; ++i) {
        int c = c4 + i;
        int gr1 = ti * 32 + r, gc1 = tj * 32 + c;
        int gr2 = tj * 32 + r, gc2 = ti * 32 + c;
        float o1 = alpha * (T1[r][c] + T2[c][r]) + ((gr1 == gc1) ? beta : 0.f);
        float o2 = alpha * (T2[r][c] + T1[c][r]) + ((gr2 == gc2) ? beta : 0.f);
        M[(size_t)gr1 * NN + gc1] = o1;
        M[(size_t)gr2 * NN + gc2] = o2;
    }
}

// ---------------------------------------------------------------------------
// K4: assemble dense Q (coalesced gather; all transposes already folded in).
// ---------------------------------------------------------------------------
__global__ void assemble_kernel(const float* __restrict__ ws, float* __restrict__ out) {
    unsigned gid = blockIdx.x * blockDim.x + threadIdx.x;
    const unsigned total = (unsigned)NB * QDIM * QDIM;     // 52,428,800
    if (gid >= total) return;
    unsigned per = (unsigned)QDIM * QDIM;
    int b = gid / per;
    unsigned rem = gid - (unsigned)b * per;
    int r = rem / QDIM, c = rem % QDIM;
    int i = r >> 10, j = c >> 10;
    int ri = r & 1023, ci = c & 1023;

    float q = 0.f;
    int d = i - j;
    if (d == 0) {
        if (i == 0) {
            q = ws[WS_G + (size_t)b * MATS + ((size_t)ri << 10) + ci];
            if (ri == ci) q += 1.05f;
        } else {
            q = ws[WS_M2 + (size_t)(b * 4 + i - 1) * MATS + ((size_t)ri << 10) + ci];
            if (ri == ci) q += (i < N_T - 1) ? 1.05f : 0.05f;
        }
    } else if (d == 1 || d == -1) {
        int t = (d == 1) ? j : i;                          // min(i,j); E block is symmetric
        q = ws[WS_A + (size_t)(b * 4 + t) * MATS + ((size_t)ri << 10) + ci];
    }
    out[gid] = q;
}

// ---------------------------------------------------------------------------
extern "C" void kernel_launch(void* const* d_in, const int* in_sizes, int n_in,
                              void* d_out, int out_size, void* d_ws, size_t ws_size,
                              hipStream_t stream) {
    (void)in_sizes; (void)n_in; (void)out_size; (void)ws_size;
    const float* params = (const float*)d_in[0];
    float* out = (float*)d_out;
    float* ws  = (float*)d_ws;

    build_coef_kernel<<<(NMAT * NN + 255) / 256, 256, 0, stream>>>(params, ws + WS_COEF);

    unsigned nA = (unsigned)NMAT * MATS;
    build_A_kernel<<<(nA + 255) / 256, 256, 0, stream>>>(ws + WS_COEF, ws + WS_A);

    dim3 g2(NN / BM, NN / BN, NMAT + NB);                  // 8 x 8 x 10
    gemm_wmma_kernel<<<g2, 256, 0, stream>>>(ws);

    sym_kernel<<<dim3(528, NMAT), 256, 0, stream>>>(ws + WS_M2,  0.5f,  0.f);
    sym_kernel<<<dim3(528, NMAT), 256, 0, stream>>>(ws + WS_A,  -0.5f, -1.f);

    unsigned total = (unsigned)NB * QDIM * QDIM;
    assemble_kernel<<<(total + 255) / 256, 256, 0, stream>>>(ws, out);
}